// Isolator_10582799417841
// MI455X (gfx1250) — compile-verified
//
#include <hip/hip_runtime.h>
#include <math.h>

// ---------------------------------------------------------------------------
// Isolator pipeline for MI455X (gfx1250, wave32).
//   - 7 residual conv1d layers w/ batch-stat BN (conv -> stats -> apply)
//   - 1x1 conv + sigmoid -> band params (flat reshape semantics of reference)
//   - per-32-sample-block FIR synthesis + apply via V_WMMA_F32_16X16X4_F32
// All buffers are tiny (<3 MB) -> L2 resident; f32 kept for FIR accuracy.
// Filter matrix is padded to 16 WMMA columns in LDS so the inner WMMA loop
// has no EXEC-divergent loads (pure ds_load + v_wmma).
// ---------------------------------------------------------------------------

typedef __attribute__((ext_vector_type(2))) float v2f;
typedef __attribute__((ext_vector_type(8))) float v8f;

#define PI_F 3.14159265358979323846f

static inline int cdiv(int a, int b) { return (a + b - 1) / b; }

// ------------------------- generic conv1d ----------------------------------
__global__ void conv1d_k(const float* __restrict__ in, const float* __restrict__ w,
                         const float* __restrict__ bias, float* __restrict__ out,
                         int B, int Cin, int Cout, int Lin, int Lout,
                         int stride, int dil, int pad, int Kw)
{
    int idx = blockIdx.x * blockDim.x + threadIdx.x;
    int total = B * Cout * Lout;
    if (idx >= total) return;
    int t  = idx % Lout;
    int co = (idx / Lout) % Cout;
    int b  = idx / (Lout * Cout);
    float acc = bias ? bias[co] : 0.0f;
    const float* ibase = in + (size_t)b * Cin * Lin;
    const float* wbase = w + (size_t)co * Cin * Kw;
    for (int ci = 0; ci < Cin; ++ci) {
        const float* ip = ibase + ci * Lin;
        const float* wp = wbase + ci * Kw;
        for (int k = 0; k < Kw; ++k) {
            int pos = t * stride + k * dil - pad;
            float xv = (pos >= 0 && pos < Lin) ? ip[pos] : 0.0f;
            acc = fmaf(wp[k], xv, acc);
        }
    }
    out[idx] = acc;
}

// -------------------- per-channel mean / rstd over (B, L) ------------------
__global__ void bn_stats_k(const float* __restrict__ x, float* __restrict__ stats,
                           int B, int C, int L)
{
    int c = blockIdx.x;
    int n = B * L;
    float s = 0.0f, ss = 0.0f;
    for (int i = threadIdx.x; i < n; i += blockDim.x) {
        int b = i / L, t = i - b * L;
        float v = x[((size_t)b * C + c) * L + t];
        s += v; ss += v * v;
    }
    __shared__ float sh_s[256];
    __shared__ float sh_q[256];
    sh_s[threadIdx.x] = s; sh_q[threadIdx.x] = ss;
    __syncthreads();
    for (int off = blockDim.x >> 1; off > 0; off >>= 1) {
        if ((int)threadIdx.x < off) {
            sh_s[threadIdx.x] += sh_s[threadIdx.x + off];
            sh_q[threadIdx.x] += sh_q[threadIdx.x + off];
        }
        __syncthreads();
    }
    if (threadIdx.x == 0) {
        float inv_n = 1.0f / (float)n;
        float mean  = sh_s[0] * inv_n;
        float var   = sh_q[0] * inv_n - mean * mean;
        stats[2 * c]     = mean;
        stats[2 * c + 1] = rsqrtf(var + 1e-5f);
    }
}

// -------------------------- BN + ReLU (in place) ---------------------------
__global__ void bn_relu_k(float* __restrict__ x, const float* __restrict__ stats,
                          const float* __restrict__ g, const float* __restrict__ bt,
                          int B, int C, int L)
{
    int idx = blockIdx.x * blockDim.x + threadIdx.x;
    if (idx >= B * C * L) return;
    int c = (idx / L) % C;
    float v = (x[idx] - stats[2 * c]) * stats[2 * c + 1] * g[c] + bt[c];
    x[idx] = fmaxf(v, 0.0f);
}

// ------------- fused: BN(conv2) + [BN(ds) | identity] -> ReLU --------------
__global__ void res_add_k(const float* __restrict__ h2, const float* __restrict__ st2,
                          const float* __restrict__ g2, const float* __restrict__ b2,
                          const float* __restrict__ dsraw, const float* __restrict__ dst,
                          const float* __restrict__ dg, const float* __restrict__ db,
                          const float* __restrict__ xid, float* __restrict__ out,
                          int B, int C, int L)
{
    int idx = blockIdx.x * blockDim.x + threadIdx.x;
    if (idx >= B * C * L) return;
    int c = (idx / L) % C;
    float v = (h2[idx] - st2[2 * c]) * st2[2 * c + 1] * g2[c] + b2[c];
    float r;
    if (dsraw) r = (dsraw[idx] - dst[2 * c]) * dst[2 * c + 1] * dg[c] + db[c];
    else       r = xid[idx];
    out[idx] = fmaxf(v + r, 0.0f);
}

// --------------------- 1x1 conv (32->12) + sigmoid -------------------------
__global__ void reg_sig_k(const float* __restrict__ in, const float* __restrict__ w,
                          const float* __restrict__ bias, float* __restrict__ y,
                          int B, int Cin, int Cout, int L)
{
    int idx = blockIdx.x * blockDim.x + threadIdx.x;
    if (idx >= B * Cout * L) return;
    int tt = idx % L;
    int ch = (idx / L) % Cout;
    int b  = idx / (L * Cout);
    float s = bias[ch];
    const float* ib = in + (size_t)b * Cin * L;
    for (int ci = 0; ci < Cin; ++ci)
        s = fmaf(w[ch * Cin + ci], ib[ci * L + tt], s);
    y[idx] = 1.0f / (1.0f + expf(-s));
}

// -------- extract lo/hi/vol with the reference's flat-reshape indexing -----
__global__ void lohivol_k(const float* __restrict__ y, float* __restrict__ lohv,
                          int B, int Tp, int tot /* = B*Tp*4 */)
{
    int idx = blockIdx.x * blockDim.x + threadIdx.x;
    if (idx >= tot) return;
    int n = idx & 3;
    int t = (idx >> 2) % Tp;
    int b = idx / (4 * Tp);
    const float* yb = y + (size_t)b * 12 * Tp;   // flat (12*Tp) per batch
    int f0 = t * 12 + n * 3;
    float r0 = yb[f0], r1 = yb[f0 + 1], r2 = yb[f0 + 2];
    float lo = r0 * 0.5f;
    float hi = lo + r1 * (0.5f - lo);
    lohv[idx]           = lo;
    lohv[tot + idx]     = hi;
    lohv[2 * tot + idx] = r2;   // vol
}

// ------------------------- FIR apply via WMMA ------------------------------
// One wave per (b, tp) block. D[32x16] = W[32x256] * F[256x16] with
// V_WMMA_F32_16X16X4_F32; K padded 255->256 with a zero tap; F columns 4..15
// are zero so only N=0..3 carry real targets (branchless inner loop).
__device__ inline void store_iso(float* __restrict__ out, const float* __restrict__ vol,
                                 int b, int n, int t, float iso)
{
    // volume lerp to full rate: src = (t+0.5)/32 - 0.5, clipped to [0, 2047]
    float src = ((float)t + 0.5f) * (1.0f / 32.0f) - 0.5f;
    src = fminf(fmaxf(src, 0.0f), 2047.0f);
    int   i0 = (int)floorf(src);
    int   i1 = (i0 + 1 > 2047) ? 2047 : i0 + 1;
    float w1 = src - (float)i0;
    float v0 = vol[((size_t)b * 2048 + i0) * 4 + n];
    float v1 = vol[((size_t)b * 2048 + i1) * 4 + n];
    float vf = v0 * (1.0f - w1) + v1 * w1;
    out[((size_t)b * 4 + n) * 65536 + t] = iso * vf;
}

__global__ __launch_bounds__(32) void iso_wmma_k(const float* __restrict__ x,
                                                 const float* __restrict__ lohv,
                                                 float* __restrict__ out,
                                                 int tot /* = B*2048*4 */)
{
    const int T = 65536, Tp = 2048;
    int blk = blockIdx.x;        // b*Tp + tp
    int tp  = blk % Tp;
    int b   = blk / Tp;
    int lane = threadIdx.x;

    __shared__ float xs[320];        // window samples (need 287)
    __shared__ float F[256 * 16];    // filter taps [k][n], padded to 16 cols

    // per-target band edges (x2 as in reference)
    float lo2[4], hi2[4];
    #pragma unroll
    for (int n = 0; n < 4; ++n) {
        int li = (b * Tp + tp) * 4 + n;
        lo2[n] = 2.0f * lohv[li];
        hi2[n] = 2.0f * lohv[tot + li];
    }

    // stage signal window: x[b, tp*32-127 .. tp*32+160], zero outside [0,T)
    int base = tp * 32 - 127;
    for (int j = lane; j < 288; j += 32) {
        int pos = base + j;
        xs[j] = (pos >= 0 && pos < T) ? x[(size_t)b * T + pos] : 0.0f;
    }

    // synthesize band-pass FIR: (hi2*sinc(hi2*m) - lo2*sinc(lo2*m)) * hamming
    for (int k = lane; k < 256; k += 32) {
        float* Frow = &F[k * 16];
        #pragma unroll
        for (int n = 4; n < 16; ++n) Frow[n] = 0.0f;
        if (k >= 255) {
            Frow[0] = 0.0f; Frow[1] = 0.0f; Frow[2] = 0.0f; Frow[3] = 0.0f;
            continue;
        }
        float m   = (float)k - 127.0f;
        float fac = -PI_F + (float)k * (2.0f * PI_F / 254.0f);
        float win = 0.54f + 0.46f * cosf(fac);
        #pragma unroll
        for (int n = 0; n < 4; ++n) {
            float zh = hi2[n] * m, zl = lo2[n] * m;
            float sh = (fabsf(zh) < 1e-9f) ? 1.0f : sinf(PI_F * zh) / (PI_F * zh);
            float sl = (fabsf(zl) < 1e-9f) ? 1.0f : sinf(PI_F * zl) / (PI_F * zl);
            Frow[n] = (hi2[n] * sh - lo2[n] * sl) * win;
        }
    }
    __syncthreads();

    // WMMA: A(16x4 f32): lanes 0-15 hold K=0,1; lanes 16-31 hold K=2,3 (M=lane&15)
    //       B(4x16 f32): mirrored (N=lane&15, same K split)
    //       C/D(16x16 f32): VGPR r -> rows r (lanes 0-15) and r+8 (lanes 16-31)
    v8f c0 = {};
    v8f c1 = {};
    int mrow = lane & 15;
    int koff = (lane >> 4) << 1;   // 0 for lanes 0-15, 2 for lanes 16-31
    for (int j = 0; j < 64; ++j) {
        int k0 = 4 * j + koff;
        v2f a0, a1, bb;
        a0[0] = xs[mrow + k0];       a0[1] = xs[mrow + k0 + 1];
        a1[0] = xs[16 + mrow + k0];  a1[1] = xs[16 + mrow + k0 + 1];
        bb[0] = F[k0 * 16 + mrow];   bb[1] = F[(k0 + 1) * 16 + mrow];
        c0 = __builtin_amdgcn_wmma_f32_16x16x4_f32(false, a0, false, bb,
                                                   (short)0, c0, false, false);
        c1 = __builtin_amdgcn_wmma_f32_16x16x4_f32(false, a1, false, bb,
                                                   (short)0, c1, false, false);
    }

    // scatter D, scaling by interpolated volume
    int n = lane & 15;
    if (n < 4) {
        const float* vol = lohv + 2 * tot;
        int mbase = (lane >> 4) * 8;
        #pragma unroll
        for (int r = 0; r < 8; ++r) {
            int m0 = mbase + r;
            store_iso(out, vol, b, n, tp * 32 + m0,      c0[r]);
            store_iso(out, vol, b, n, tp * 32 + 16 + m0, c1[r]);
        }
    }
}

// ---------------------------------------------------------------------------
// Host orchestration
// ---------------------------------------------------------------------------
struct LayerCfg { int cin, cout, Lin, Lout, stride, dil, ds; };
static const LayerCfg LC[7] = {
    { 1,  2, 65536, 32768, 2, 1, 1},
    { 2,  4, 32768, 16384, 2, 2, 1},
    { 4,  4, 16384, 16384, 1, 2, 0},
    { 4,  8, 16384,  8192, 2, 1, 1},
    { 8,  8,  8192,  4096, 2, 2, 1},
    { 8, 16,  4096,  4096, 1, 2, 1},
    {16, 32,  4096,  2048, 2, 1, 1},
};

extern "C" void kernel_launch(void* const* d_in, const int* in_sizes, int n_in,
                              void* d_out, int out_size, void* d_ws, size_t ws_size,
                              hipStream_t stream)
{
    (void)in_sizes; (void)n_in; (void)out_size; (void)ws_size;
    const int B = 2;
    const float* x = (const float*)d_in[0];

    // workspace carve-out (floats)
    float* ws     = (float*)d_ws;
    float* bufA   = ws;                 // 131072
    float* bufB   = bufA   + 131072;    // 131072
    float* t1     = bufB   + 131072;    // conv1 out / h1 (in place)
    float* t2     = t1     + 131072;    // ds conv out
    float* t3     = t2     + 131072;    // conv2 out
    float* stats1 = t3     + 131072;    // 2*C interleaved (mean, rstd)
    float* stats2 = stats1 + 128;
    float* stats3 = stats2 + 128;
    float* yflat  = stats3 + 128;       // B*12*2048
    float* lohv   = yflat  + 49152;     // 3 planes of B*2048*4

    // params (JAX pytree order: dict keys sorted alphabetically)
    int pi = 1;
    const float* cur = x;
    for (int i = 0; i < 7; ++i) {
        const LayerCfg& L = LC[i];
        const float* bn1_b   = (const float*)d_in[pi++];
        const float* bn1_g   = (const float*)d_in[pi++];
        const float* bn2_b   = (const float*)d_in[pi++];
        const float* bn2_g   = (const float*)d_in[pi++];
        const float* conv1_b = (const float*)d_in[pi++];
        const float* conv1_w = (const float*)d_in[pi++];
        const float* conv2_b = (const float*)d_in[pi++];
        const float* conv2_w = (const float*)d_in[pi++];
        const float* ds_b = nullptr, *ds_g = nullptr, *ds_w = nullptr;
        if (L.ds) {
            ds_b = (const float*)d_in[pi++];
            ds_g = (const float*)d_in[pi++];
            ds_w = (const float*)d_in[pi++];
        }

        int tot = B * L.cout * L.Lout;
        int pad = 2 * L.dil;

        // h1 = relu(bn1(conv1(x)))
        conv1d_k<<<cdiv(tot, 256), 256, 0, stream>>>(cur, conv1_w, conv1_b, t1,
            B, L.cin, L.cout, L.Lin, L.Lout, L.stride, L.dil, pad, 5);
        bn_stats_k<<<L.cout, 256, 0, stream>>>(t1, stats1, B, L.cout, L.Lout);
        bn_relu_k<<<cdiv(tot, 256), 256, 0, stream>>>(t1, stats1, bn1_g, bn1_b,
            B, L.cout, L.Lout);

        // conv2(h1) (raw) + stats
        conv1d_k<<<cdiv(tot, 256), 256, 0, stream>>>(t1, conv2_w, conv2_b, t3,
            B, L.cout, L.cout, L.Lout, L.Lout, 1, L.dil, pad, 5);
        bn_stats_k<<<L.cout, 256, 0, stream>>>(t3, stats3, B, L.cout, L.Lout);

        // downsample path (raw) + stats
        if (L.ds) {
            conv1d_k<<<cdiv(tot, 256), 256, 0, stream>>>(cur, ds_w, nullptr, t2,
                B, L.cin, L.cout, L.Lin, L.Lout, L.stride, 1, 0, 1);
            bn_stats_k<<<L.cout, 256, 0, stream>>>(t2, stats2, B, L.cout, L.Lout);
        }

        // out = relu(bn2(conv2) + residual)
        float* dst = (cur == bufA) ? bufB : bufA;
        res_add_k<<<cdiv(tot, 256), 256, 0, stream>>>(t3, stats3, bn2_g, bn2_b,
            L.ds ? t2 : nullptr, stats2, ds_g, ds_b, cur, dst,
            B, L.cout, L.Lout);
        cur = dst;
    }

    const float* reg_b = (const float*)d_in[pi++];
    const float* reg_w = (const float*)d_in[pi++];

    // band-parameter head
    reg_sig_k<<<cdiv(B * 12 * 2048, 256), 256, 0, stream>>>(cur, reg_w, reg_b,
        yflat, B, 32, 12, 2048);
    int tot4 = B * 2048 * 4;
    lohivol_k<<<cdiv(tot4, 256), 256, 0, stream>>>(yflat, lohv, B, 2048, tot4);

    // FIR synthesis + apply + volume scale (WMMA)
    iso_wmma_k<<<B * 2048, 32, 0, stream>>>(x, lohv, (float*)d_out, tot4);
}